// RelGraphConvLayer_1331439862167
// MI455X (gfx1250) — compile-verified
//
#include <hip/hip_runtime.h>
#include <hip/hip_bf16.h>

typedef __attribute__((ext_vector_type(2))) float v2f;
typedef __attribute__((ext_vector_type(8))) float v8f;

#define FEAT 128

// ---------------------------------------------------------------------------
// Scatter phase: one wave32 per edge. 32 lanes x float4 = 128-feature row.
// lane 0 bumps the degree counter; all lanes gather the source row and
// atomically accumulate into the destination aggregate (L2-resident).
// Relations: y==0 follows (user->user), y==1 bought_by (item->user),
//            y==2 buys (user->item)
// ---------------------------------------------------------------------------
__global__ __launch_bounds__(256) void scatter_agg_kernel(
    const float* __restrict__ x_user, const float* __restrict__ x_item,
    const long long* __restrict__ e_follows,
    const long long* __restrict__ e_buys,
    const long long* __restrict__ e_bought_by,
    float* __restrict__ agg_f, float* __restrict__ agg_bb, float* __restrict__ agg_b,
    float* __restrict__ deg_f, float* __restrict__ deg_bb, float* __restrict__ deg_b,
    int n_edges)
{
    const int r = blockIdx.y;
    const long long* E;
    const float* X;
    float* AGG;
    float* DEG;
    if (r == 0)      { E = e_follows;   X = x_user; AGG = agg_f;  DEG = deg_f;  }
    else if (r == 1) { E = e_bought_by; X = x_item; AGG = agg_bb; DEG = deg_bb; }
    else             { E = e_buys;      X = x_user; AGG = agg_b;  DEG = deg_b;  }

    const int edge = blockIdx.x * 8 + (threadIdx.x >> 5);
    if (edge >= n_edges) return;
    const int lane = threadIdx.x & 31;

    const long long s = E[edge];
    const long long d = E[n_edges + edge];

    if (lane == 0) unsafeAtomicAdd(DEG + d, 1.0f);

    const float4 v = *(const float4*)(X + (size_t)s * FEAT + lane * 4);
    float* p = AGG + (size_t)d * FEAT + lane * 4;
    unsafeAtomicAdd(p + 0, v.x);
    unsafeAtomicAdd(p + 1, v.y);
    unsafeAtomicAdd(p + 2, v.z);
    unsafeAtomicAdd(p + 3, v.w);
}

// ---------------------------------------------------------------------------
// Fused epilogue GEMM: out[m,:] = relu( sum_s (scale_s(A_s[m,:])) @ W_s + bias )
// One block = one 16-row M tile; 8 waves cover the 8 N tiles of 16 columns.
// Each source's 16x128 A tile is staged once in LDS (degree scaling applied
// during the stage), then consumed by v_wmma_f32_16x16x4_f32 over K.
// ---------------------------------------------------------------------------
#define LDS_PITCH 132   // 128 + 4 pad: row r starts at bank 4r -> conflict-free

__global__ __launch_bounds__(256) void fused_wmma_gemm_kernel(
    const float* __restrict__ A0, const float* __restrict__ D0, const float* __restrict__ W0,
    const float* __restrict__ A1, const float* __restrict__ D1, const float* __restrict__ W1,
    const float* __restrict__ A2, const float* __restrict__ D2, const float* __restrict__ W2,
    const float* __restrict__ bias, float* __restrict__ out)
{
    __shared__ float As[16 * LDS_PITCH];

    const int mbase = blockIdx.x * 16;
    const int tid   = threadIdx.x;
    const int wave  = tid >> 5;      // 0..7 -> N tile
    const int lane  = tid & 31;
    const int half  = lane >> 4;     // 0/1 -> K sub-pair (and M half for C/D)
    const int l16   = lane & 15;
    const int ncol  = wave * 16 + l16;

    v8f acc = {};

    const float* Asrc[3] = {A0, A1, A2};
    const float* Dsrc[3] = {D0, D1, D2};
    const float* Wsrc[3] = {W0, W1, W2};

#pragma unroll
    for (int s = 0; s < 3; ++s) {
        const float* A = Asrc[s];
        if (A == nullptr) break;                 // uniform branch
        const float* W = Wsrc[s];
        const float* DEG = Dsrc[s];

        __syncthreads();                         // protect LDS from prior reads
        // stage 16x128 A tile (2048 floats, 8 per thread), fold in 1/deg
        for (int i = tid; i < 16 * FEAT; i += 256) {
            const int row = i >> 7;
            const int col = i & (FEAT - 1);
            float v = A[(size_t)(mbase + row) * FEAT + col];
            if (DEG != nullptr) {
                const float dg = DEG[mbase + row];
                v *= 1.0f / fmaxf(dg, 1.0f);
            }
            As[row * LDS_PITCH + col] = v;
        }
        __syncthreads();

#pragma unroll 4
        for (int k0 = 0; k0 < FEAT; k0 += 4) {
            const int kk = k0 + 2 * half;
            v2f a, b;
            // A 16x4 frag: lane<16 -> K={k0,k0+1}, lane>=16 -> K={k0+2,k0+3}
            a.x = As[l16 * LDS_PITCH + kk];
            a.y = As[l16 * LDS_PITCH + kk + 1];
            // B 4x16 frag: same K split, N = lane%16 ; W is [K=128][N=128]
            b.x = W[(size_t)kk * FEAT + ncol];
            b.y = W[(size_t)(kk + 1) * FEAT + ncol];
            acc = __builtin_amdgcn_wmma_f32_16x16x4_f32(
                false, a, false, b, (short)0, acc, false, false);
        }
    }

    // bias + ReLU + store: D[v + 8*half][l16]
    const float bv = bias[ncol];
#pragma unroll
    for (int v = 0; v < 8; ++v) {
        const int row = mbase + v + 8 * half;
        out[(size_t)row * FEAT + ncol] = fmaxf(acc[v] + bv, 0.0f);
    }
}

// ---------------------------------------------------------------------------
extern "C" void kernel_launch(void* const* d_in, const int* in_sizes, int n_in,
                              void* d_out, int out_size, void* d_ws, size_t ws_size,
                              hipStream_t stream) {
    const float*     x_user    = (const float*)d_in[0];
    const float*     x_item    = (const float*)d_in[1];
    const long long* e_follows = (const long long*)d_in[2];
    const long long* e_buys    = (const long long*)d_in[3];
    const long long* e_bought  = (const long long*)d_in[4];
    const float*     W_follows = (const float*)d_in[5];
    const float*     W_buys    = (const float*)d_in[6];
    const float*     W_bought  = (const float*)d_in[7];
    const float*     W_loop_u  = (const float*)d_in[8];
    const float*     b_loop_u  = (const float*)d_in[9];
    const float*     W_loop_i  = (const float*)d_in[10];
    const float*     b_loop_i  = (const float*)d_in[11];
    float* out = (float*)d_out;

    const int NU = in_sizes[0] / FEAT;          // 50000
    const int NI = in_sizes[1] / FEAT;          // 50000
    const int NE = in_sizes[2] / 2;             // 600000

    float* ws      = (float*)d_ws;
    float* agg_f   = ws;                               // [NU*128] follows -> user
    float* agg_bb  = ws + (size_t)NU * FEAT;           // [NU*128] bought_by -> user
    float* agg_b   = ws + 2 * (size_t)NU * FEAT;       // [NI*128] buys -> item
    float* deg_f   = ws + 2 * (size_t)NU * FEAT + (size_t)NI * FEAT;
    float* deg_bb  = deg_f + NU;
    float* deg_b   = deg_bb + NU;

    size_t need = (2 * (size_t)NU * FEAT + (size_t)NI * FEAT
                   + 2 * (size_t)NU + (size_t)NI) * sizeof(float);
    if (need > ws_size) need = ws_size;
    hipMemsetAsync(d_ws, 0, need, stream);

    dim3 sgrid((NE + 7) / 8, 3);
    scatter_agg_kernel<<<sgrid, 256, 0, stream>>>(
        x_user, x_item, e_follows, e_buys, e_bought,
        agg_f, agg_bb, agg_b, deg_f, deg_bb, deg_b, NE);

    // h_user = relu(agg_f@Wf + agg_bb@Wbb + x_user@Wlu + b_lu)
    fused_wmma_gemm_kernel<<<NU / 16, 256, 0, stream>>>(
        agg_f, deg_f, W_follows,
        agg_bb, deg_bb, W_bought,
        x_user, nullptr, W_loop_u,
        b_loop_u, out);

    // h_item = relu(agg_b@Wb + x_item@Wli + b_li)
    fused_wmma_gemm_kernel<<<NI / 16, 256, 0, stream>>>(
        agg_b, deg_b, W_buys,
        x_item, nullptr, W_loop_i,
        nullptr, nullptr, nullptr,
        b_loop_i, out + (size_t)NU * FEAT);
}